// AttentionEncoder_51075751084120
// MI455X (gfx1250) — compile-verified
//
#include <hip/hip_runtime.h>
#include <math.h>

#define T_DIM 1024
#define B_DIM 64
#define D_DIM 256
#define L_DIM 1024
#define DC_DIM 256
#define H_DIM 256
#define A_DIM 256
#define G3 (3 * H_DIM)
#define NEG_INF (-1e9f)

typedef float v2f __attribute__((ext_vector_type(2)));
typedef float v8f __attribute__((ext_vector_type(8)));

// ---------------------------------------------------------------------------
// Prelude GEMM: C[M,N] = A[M,K] @ W[K,N], fp32 via V_WMMA_F32_16X16X4_F32.
// Grid: (M/128, N/128), 256 threads = 8 waves; wave w owns rows [m0+16w,+16)
// and all eight 16-col tiles of the 128-col block (A frag reused 8x per k).
// ---------------------------------------------------------------------------
__global__ __launch_bounds__(256) void gemm_wmma_f32(
    const float* __restrict__ Am, const float* __restrict__ Wm,
    float* __restrict__ Cm, int M, int N, int K) {
  const int lane = threadIdx.x & 31;
  const int wave = threadIdx.x >> 5;
  const int halfsel = lane >> 4;   // 0: lanes 0-15, 1: lanes 16-31
  const int l15 = lane & 15;
  const int m0 = blockIdx.x * 128 + wave * 16;
  const int n0 = blockIdx.y * 128;

  v8f acc[8];
#pragma unroll
  for (int i = 0; i < 8; ++i)
#pragma unroll
    for (int r = 0; r < 8; ++r) acc[i][r] = 0.0f;

  const size_t arow = (size_t)(m0 + l15) * K;
  for (int k0 = 0; k0 < K; k0 += 4) {
    const int ka = k0 + (halfsel << 1);          // K for .x ; .y is ka+1
    v2f a;
    a.x = Am[arow + ka];
    a.y = Am[arow + ka + 1];
#pragma unroll
    for (int nt = 0; nt < 8; ++nt) {
      const int col = n0 + nt * 16 + l15;
      v2f b;
      b.x = Wm[(size_t)ka * N + col];
      b.y = Wm[(size_t)(ka + 1) * N + col];
      acc[nt] = __builtin_amdgcn_wmma_f32_16x16x4_f32(
          false, a, false, b, (short)0, acc[nt], false, false);
    }
  }
#pragma unroll
  for (int nt = 0; nt < 8; ++nt) {
    const int col = n0 + nt * 16 + l15;
#pragma unroll
    for (int r = 0; r < 8; ++r) {
      const int row = m0 + r + (halfsel << 3);
      Cm[(size_t)row * N + col] = acc[nt][r];
    }
  }
}

// ---------------------------------------------------------------------------
// Init: h = 0, q = xproj[t=0] (h0 == 0), gh = b_hh broadcast.
// ---------------------------------------------------------------------------
__global__ void init_kernel(const float* __restrict__ xproj,
                            const float* __restrict__ b_hh,
                            float* __restrict__ h, float* __restrict__ q,
                            float* __restrict__ gh) {
  const int i = blockIdx.x * blockDim.x + threadIdx.x;
  if (i < B_DIM * H_DIM) h[i] = 0.0f;
  if (i < B_DIM * A_DIM) q[i] = xproj[i];
  if (i < B_DIM * G3) gh[i] = b_hh[i % G3];
}

__device__ __forceinline__ float wave_sum(float v) {
#pragma unroll
  for (int off = 16; off > 0; off >>= 1) v += __shfl_xor(v, off, 32);
  return v;
}
__device__ __forceinline__ float wave_max(float v) {
#pragma unroll
  for (int off = 16; off > 0; off >>= 1) v = fmaxf(v, __shfl_xor(v, off, 32));
  return v;
}

// ---------------------------------------------------------------------------
// scores[b,l] = masked( dot(tanh(ctx_proj[b,l,:] + q[b,:]), v) )
// Grid (8, B); each of 8 waves owns 16 l-rows -> no cross-wave sync.
// Lane owns contiguous a = [8*lane, 8*lane+8): two b128 loads per row,
// wave still covers 1024 contiguous bytes (fully coalesced).
// ---------------------------------------------------------------------------
__global__ __launch_bounds__(256) void scores_kernel(
    const float* __restrict__ ctxp, const float* __restrict__ q,
    const float* __restrict__ vvec, const int* __restrict__ ctx_len,
    float* __restrict__ scores) {
  const int b = blockIdx.y;
  const int wave = threadIdx.x >> 5;
  const int lane = threadIdx.x & 31;
  const int lbase = blockIdx.x * 128 + wave * 16;
  const int cl = ctx_len[b];

  const float4* qp = reinterpret_cast<const float4*>(q + b * A_DIM) + lane * 2;
  const float4* vp = reinterpret_cast<const float4*>(vvec) + lane * 2;
  const float4 q0 = qp[0], q1 = qp[1];
  const float4 v0 = vp[0], v1 = vp[1];

  for (int i = 0; i < 16; ++i) {
    const int l = lbase + i;
    const float4* row =
        reinterpret_cast<const float4*>(ctxp + ((size_t)b * L_DIM + l) * A_DIM) +
        lane * 2;
    const float4 r0 = row[0];
    const float4 r1 = row[1];
    float s = tanhf(r0.x + q0.x) * v0.x;
    s += tanhf(r0.y + q0.y) * v0.y;
    s += tanhf(r0.z + q0.z) * v0.z;
    s += tanhf(r0.w + q0.w) * v0.w;
    s += tanhf(r1.x + q1.x) * v1.x;
    s += tanhf(r1.y + q1.y) * v1.y;
    s += tanhf(r1.z + q1.z) * v1.z;
    s += tanhf(r1.w + q1.w) * v1.w;
    s = wave_sum(s);
    if (lane == 0) scores[b * L_DIM + l] = (l < cl) ? s : NEG_INF;
  }
}

// ---------------------------------------------------------------------------
// Per-batch: softmax over L, c = sum_l alpha*context[b,l,:], then
// gi = x_t @ W_ih[:D] + c @ W_ih[D:] + b_ih (thread owns 3 output cols).
// ---------------------------------------------------------------------------
__global__ __launch_bounds__(256) void softmax_gi_kernel(
    const float* __restrict__ scores, const float* __restrict__ context,
    const float* __restrict__ x, const float* __restrict__ W_ih,
    const float* __restrict__ b_ih, float* __restrict__ gi, int t) {
  __shared__ float alpha[L_DIM];
  __shared__ float red[8];
  __shared__ float cbuf[DC_DIM];
  __shared__ float xbuf[D_DIM];
  const int b = blockIdx.x;
  const int tid = threadIdx.x;
  const int lane = tid & 31, wave = tid >> 5;

  float s4[4];
  float m = NEG_INF;
#pragma unroll
  for (int j = 0; j < 4; ++j) {
    s4[j] = scores[b * L_DIM + tid * 4 + j];
    m = fmaxf(m, s4[j]);
  }
  m = wave_max(m);
  if (lane == 0) red[wave] = m;
  __syncthreads();
  float tmp = red[lane & 7];
  tmp = fmaxf(tmp, __shfl_xor(tmp, 4, 32));
  tmp = fmaxf(tmp, __shfl_xor(tmp, 2, 32));
  tmp = fmaxf(tmp, __shfl_xor(tmp, 1, 32));
  const float mAll = tmp;
  __syncthreads();  // done reading red before reuse

  float sum = 0.0f;
#pragma unroll
  for (int j = 0; j < 4; ++j) {
    const float e = __expf(s4[j] - mAll);
    alpha[tid * 4 + j] = e;   // unnormalized; fold 1/sum later
    sum += e;
  }
  sum = wave_sum(sum);
  if (lane == 0) red[wave] = sum;
  __syncthreads();  // also makes alpha[] visible
  tmp = red[lane & 7];
  tmp += __shfl_xor(tmp, 4, 32);
  tmp += __shfl_xor(tmp, 2, 32);
  tmp += __shfl_xor(tmp, 1, 32);
  const float inv = 1.0f / tmp;

  // c[b, tid] = inv * sum_l alpha[l] * context[b,l,tid]  (coalesced rows)
  float c = 0.0f;
  const float* cb = context + (size_t)b * L_DIM * DC_DIM + tid;
  for (int l = 0; l < L_DIM; ++l) c += alpha[l] * cb[(size_t)l * DC_DIM];
  cbuf[tid] = c * inv;
  xbuf[tid] = x[((size_t)t * B_DIM + b) * D_DIM + tid];
  __syncthreads();

#pragma unroll
  for (int g = 0; g < 3; ++g) {
    const int col = g * H_DIM + tid;
    float acc = b_ih[col];
    for (int k = 0; k < D_DIM; ++k) acc += xbuf[k] * W_ih[(size_t)k * G3 + col];
    for (int k = 0; k < DC_DIM; ++k)
      acc += cbuf[k] * W_ih[(size_t)(D_DIM + k) * G3 + col];
    gi[b * G3 + col] = acc;
  }
}

// ---------------------------------------------------------------------------
// GRU update for step t, then (fused) next-step projections
//   q_{t+1} = xproj[t+1] + h@Wh ,  gh_{t+1} = b_hh + h@W_hh
// ---------------------------------------------------------------------------
__global__ __launch_bounds__(256) void gru_h_kernel(
    const float* __restrict__ gi, const float* __restrict__ gh,
    const float* __restrict__ xproj, const float* __restrict__ Wh,
    const float* __restrict__ W_hh, const float* __restrict__ b_hh,
    const int* __restrict__ lengths, float* __restrict__ h,
    float* __restrict__ q, float* __restrict__ gh_out,
    float* __restrict__ out, int t) {
  __shared__ float hbuf[H_DIM];
  const int b = blockIdx.x;
  const int j = threadIdx.x;

  const float gir = gi[b * G3 + j];
  const float giu = gi[b * G3 + H_DIM + j];
  const float gin = gi[b * G3 + 2 * H_DIM + j];
  const float ghr = gh[b * G3 + j];
  const float ghu = gh[b * G3 + H_DIM + j];
  const float ghn = gh[b * G3 + 2 * H_DIM + j];

  const float r = 1.0f / (1.0f + __expf(-(gir + ghr)));
  const float u = 1.0f / (1.0f + __expf(-(giu + ghu)));
  const float n = tanhf(gin + r * ghn);
  const float hold = h[b * H_DIM + j];
  const float hnew = u * hold + (1.0f - u) * n;
  const bool valid = t < lengths[b];
  const float hnext = valid ? hnew : hold;

  out[((size_t)t * B_DIM + b) * H_DIM + j] = valid ? hnew : 0.0f;
  h[b * H_DIM + j] = hnext;
  hbuf[j] = hnext;
  if (t == T_DIM - 1)
    out[(size_t)T_DIM * B_DIM * H_DIM + b * H_DIM + j] = hnext;
  __syncthreads();

  if (t + 1 < T_DIM) {
    float qa = xproj[((size_t)(t + 1) * B_DIM + b) * A_DIM + j];
    for (int k = 0; k < H_DIM; ++k) qa += hbuf[k] * Wh[(size_t)k * A_DIM + j];
    q[b * A_DIM + j] = qa;
#pragma unroll
    for (int g = 0; g < 3; ++g) {
      const int col = g * H_DIM + j;
      float acc = b_hh[col];
      for (int k = 0; k < H_DIM; ++k)
        acc += hbuf[k] * W_hh[(size_t)k * G3 + col];
      gh_out[b * G3 + col] = acc;  // safe: reads above precede barrier
    }
  }
}

extern "C" void kernel_launch(void* const* d_in, const int* in_sizes, int n_in,
                              void* d_out, int out_size, void* d_ws,
                              size_t ws_size, hipStream_t stream) {
  const float* x = (const float*)d_in[0];
  const int* lengths = (const int*)d_in[1];
  const float* context = (const float*)d_in[2];
  const int* ctx_lengths = (const int*)d_in[3];
  const float* Wc = (const float*)d_in[4];
  const float* Wx = (const float*)d_in[5];
  const float* Wh = (const float*)d_in[6];
  const float* v = (const float*)d_in[7];
  const float* W_ih = (const float*)d_in[8];
  const float* W_hh = (const float*)d_in[9];
  const float* b_ih = (const float*)d_in[10];
  const float* b_hh = (const float*)d_in[11];
  float* out = (float*)d_out;

  float* ws = (float*)d_ws;
  size_t off = 0;
  float* ctxp = ws + off;  off += (size_t)B_DIM * L_DIM * A_DIM;  // 16.78M
  float* xproj = ws + off; off += (size_t)T_DIM * B_DIM * A_DIM;  // 16.78M
  float* q = ws + off;     off += (size_t)B_DIM * A_DIM;
  float* gh = ws + off;    off += (size_t)B_DIM * G3;
  float* scores = ws + off; off += (size_t)B_DIM * L_DIM;
  float* gi = ws + off;    off += (size_t)B_DIM * G3;
  float* h = ws + off;     off += (size_t)B_DIM * H_DIM;

  // Prelude WMMA GEMMs (L2-resident thereafter):
  // ctx_proj = context @ Wc  : [B*L, DC] @ [DC, A]
  gemm_wmma_f32<<<dim3((B_DIM * L_DIM) / 128, A_DIM / 128), 256, 0, stream>>>(
      context, Wc, ctxp, B_DIM * L_DIM, A_DIM, DC_DIM);
  // xproj = x @ Wx : [T*B, D] @ [D, A]
  gemm_wmma_f32<<<dim3((T_DIM * B_DIM) / 128, A_DIM / 128), 256, 0, stream>>>(
      x, Wx, xproj, T_DIM * B_DIM, A_DIM, D_DIM);

  init_kernel<<<(B_DIM * G3 + 255) / 256, 256, 0, stream>>>(xproj, b_hh, h, q,
                                                            gh);

  for (int t = 0; t < T_DIM; ++t) {
    scores_kernel<<<dim3(8, B_DIM), 256, 0, stream>>>(ctxp, q, v, ctx_lengths,
                                                      scores);
    softmax_gi_kernel<<<B_DIM, 256, 0, stream>>>(scores, context, x, W_ih,
                                                 b_ih, gi, t);
    gru_h_kernel<<<B_DIM, 256, 0, stream>>>(gi, gh, xproj, Wh, W_hh, b_hh,
                                            lengths, h, q, gh, out, t);
  }
}